// GNN_45131516346369
// MI455X (gfx1250) — compile-verified
//
#include <hip/hip_runtime.h>
#include <stdint.h>

// GATv2Conv(64, 1024, heads=16) + linear head, fused pipeline for gfx1250.
// Memory-bound (~22 GFLOP vs a few GB of traffic @ 23.3 TB/s): stay in f32,
// V_WMMA_F32_16X16X4_F32 for the dense GEMMs, async global->LDS staging for
// the weight tiles, CSR + register-resident aggregation (no FP atomics ->
// deterministic) for the sparse half.

#define HEADS 16
#define CH    1024
#define HC    (HEADS * CH)   // 16384
#define FIN   64
#define NCLS  20
#define NEG_SLOPE 0.2f
#define CHUNK 64             // edges staged per aggregation chunk

typedef float v2f __attribute__((ext_vector_type(2)));
typedef float v8f __attribute__((ext_vector_type(8)));

// CDNA5 async global->LDS 16B copy (ASYNCcnt-tracked, bypasses VGPRs).
__device__ __forceinline__ void async_copy_b128(void* lds_dst, const void* gsrc) {
  const unsigned lds = (unsigned)(uintptr_t)lds_dst;       // addrspace(3) offset
  const unsigned long long ga = (unsigned long long)(uintptr_t)gsrc;
  asm volatile("global_load_async_to_lds_b128 %0, %1, off"
               :: "v"(lds), "v"(ga)
               : "memory");
}
__device__ __forceinline__ void wait_async0() {
  asm volatile("s_wait_asynccnt 0" ::: "memory");
}

// ---------------------------------------------------------------------------
// Kernel 1: xl = x @ Wl ; xr = x @ Wr  (M=4096, K=64, N=16384 each)
// Block = 128 threads (4 waves); block computes a 16(M) x 256(N) tile;
// each wave owns 64 columns as four 16x16 WMMA accumulators.
// A frag (f32 16x4): lane L holds A[m][kb], A[m][kb+1], m=L&15, kb=2*(L>>4).
// B frag (f32 4x16): lane L holds B[kb][n], B[kb+1][n], n=L&15.
// C/D (f32 16x16):   VGPR v: lanes0-15 -> M=v, lanes16-31 -> M=v+8, N=lane&15.
// ---------------------------------------------------------------------------
__global__ __launch_bounds__(128)
void gemm_xw(const float* __restrict__ x, const float* __restrict__ Wl,
             const float* __restrict__ Wr, float* __restrict__ xl,
             float* __restrict__ xr) {
  __shared__ float Bs[64][256];          // 64 KB W tile, L2-hot source
  const int tid  = threadIdx.x;
  const int wave = tid >> 5;
  const int lane = tid & 31;
  const int l16  = lane & 15;
  const int kb   = (lane >> 4) * 2;

  const int m0 = blockIdx.y * 16;
  int nglob = blockIdx.x * 256;          // [0, 2*HC)
  const float* W;
  float* out;
  if (nglob < HC) { W = Wl; out = xl; }
  else            { W = Wr; out = xr; nglob -= HC; }

  // stage 64x256 B tile straight into LDS via CDNA5 async copies (no VGPR
  // round trip); overlaps with the A-fragment register loads below.
#pragma unroll
  for (int i = 0; i < 32; ++i) {
    int idx4 = i * 128 + tid;            // [0, 4096) float4s
    int k    = idx4 >> 6;
    int c4   = (idx4 & 63) * 4;
    async_copy_b128(&Bs[k][c4], W + (size_t)k * HC + nglob + c4);
  }

  v2f a[16];
  const float* xrow = x + (size_t)(m0 + l16) * FIN;
#pragma unroll
  for (int i = 0; i < 16; ++i) {
    a[i].x = xrow[kb + 4 * i];
    a[i].y = xrow[kb + 4 * i + 1];
  }
  wait_async0();                         // this wave's async copies landed
  __syncthreads();                       // tile visible to all waves

  v8f c0 = {}, c1 = {}, c2 = {}, c3 = {};
  const int nb = wave * 64 + l16;
#pragma unroll
  for (int ks = 0; ks < 16; ++ks) {
    const int k = ks * 4 + kb;
    v2f b;
    b.x = Bs[k][nb +  0]; b.y = Bs[k + 1][nb +  0];
    c0 = __builtin_amdgcn_wmma_f32_16x16x4_f32(false, a[ks], false, b,
                                               (short)0, c0, false, false);
    b.x = Bs[k][nb + 16]; b.y = Bs[k + 1][nb + 16];
    c1 = __builtin_amdgcn_wmma_f32_16x16x4_f32(false, a[ks], false, b,
                                               (short)0, c1, false, false);
    b.x = Bs[k][nb + 32]; b.y = Bs[k + 1][nb + 32];
    c2 = __builtin_amdgcn_wmma_f32_16x16x4_f32(false, a[ks], false, b,
                                               (short)0, c2, false, false);
    b.x = Bs[k][nb + 48]; b.y = Bs[k + 1][nb + 48];
    c3 = __builtin_amdgcn_wmma_f32_16x16x4_f32(false, a[ks], false, b,
                                               (short)0, c3, false, false);
  }

  const size_t colbase = (size_t)nglob + (size_t)wave * 64 + l16;
  const int mrow = m0 + (lane >> 4) * 8;
#pragma unroll
  for (int v = 0; v < 8; ++v) {
    const size_t r = (size_t)(mrow + v) * HC + colbase;
    out[r +  0] = c0[v];
    out[r + 16] = c1[v];
    out[r + 32] = c2[v];
    out[r + 48] = c3[v];
  }
}

// ---------------------------------------------------------------------------
// CSR construction (integer atomics only; final order made deterministic by
// a per-node insertion sort on edge id).
// ---------------------------------------------------------------------------
__device__ __forceinline__ int edge_dst(const long long* ei, int e, int E) {
  return (e < E) ? (int)ei[E + e] : e - E;
}
__device__ __forceinline__ int edge_src(const long long* ei, int e, int E) {
  return (e < E) ? (int)ei[e] : e - E;
}

__global__ void build_deg(const long long* __restrict__ ei, int* __restrict__ deg,
                          int E, int EA) {
  int e = blockIdx.x * blockDim.x + threadIdx.x;
  if (e >= EA) return;
  atomicAdd(&deg[edge_dst(ei, e, E)], 1);
}

// single block: exclusive scan of deg -> rowptr (+ cursor copy).
__global__ __launch_bounds__(256)
void scan_deg(const int* __restrict__ deg, int* __restrict__ rowptr,
              int* __restrict__ cursor, int N) {
  __shared__ int part[256];
  const int t = threadIdx.x;
  const int per = (N + 255) / 256;
  const int base = t * per;
  int s = 0;
  for (int i = 0; i < per; ++i) {
    int n = base + i;
    if (n < N) s += deg[n];
  }
  part[t] = s;
  __syncthreads();
  if (t == 0) {                 // 256-entry serial exclusive scan (trivial)
    int run = 0;
    for (int j = 0; j < 256; ++j) { int v = part[j]; part[j] = run; run += v; }
  }
  __syncthreads();
  int run = part[t];
  for (int i = 0; i < per; ++i) {
    int n = base + i;
    if (n < N) { rowptr[n] = run; cursor[n] = run; run += deg[n]; }
  }
}

__global__ void scatter_edges(const long long* __restrict__ ei,
                              int* __restrict__ cursor, int* __restrict__ ebuf,
                              int E, int EA) {
  int e = blockIdx.x * blockDim.x + threadIdx.x;
  if (e >= EA) return;
  int pos = atomicAdd(&cursor[edge_dst(ei, e, E)], 1);
  ebuf[pos] = e;
}

__global__ void sort_lists(const int* __restrict__ rowptr,
                           const int* __restrict__ deg, int* __restrict__ ebuf,
                           int N) {
  int n = blockIdx.x * blockDim.x + threadIdx.x;
  if (n >= N) return;
  int s = rowptr[n], d = deg[n];
  for (int i = 1; i < d; ++i) {         // ascending edge id -> deterministic
    int v = ebuf[s + i];
    int j = i - 1;
    while (j >= 0 && ebuf[s + j] > v) { ebuf[s + j + 1] = ebuf[s + j]; --j; }
    ebuf[s + j + 1] = v;
  }
}

// ---------------------------------------------------------------------------
// Kernel 2: logits[e][h] = sum_c att[h][c]*leakyrelu(xl[src][h,c]+xr[dst][h,c])
// Block per edge; thread t -> head h=t/16, lane-in-head j=t%16 -> coalesced
// 256B float4 bursts. After this, xr is dead.
// ---------------------------------------------------------------------------
__global__ __launch_bounds__(256)
void edge_logits(const float* __restrict__ xl, const float* __restrict__ xr,
                 const long long* __restrict__ ei, const float* __restrict__ att,
                 float* __restrict__ logits, int E) {
  const int e = blockIdx.x;
  const int src = edge_src(ei, e, E);
  const int dst = edge_dst(ei, e, E);

  const int t = threadIdx.x;
  const int h = t >> 4, j = t & 15;
  const float* lrow = xl + (size_t)src * HC + h * CH + j * 4;
  const float* rrow = xr + (size_t)dst * HC + h * CH + j * 4;
  const float* arow = att + h * CH + j * 4;

  float acc = 0.f;
#pragma unroll
  for (int i = 0; i < 16; ++i) {
    float4 lv = *(const float4*)(lrow + i * 64);
    float4 rv = *(const float4*)(rrow + i * 64);
    float4 av = *(const float4*)(arow + i * 64);
    float vx = lv.x + rv.x; vx = vx > 0.f ? vx : NEG_SLOPE * vx; acc += vx * av.x;
    float vy = lv.y + rv.y; vy = vy > 0.f ? vy : NEG_SLOPE * vy; acc += vy * av.y;
    float vz = lv.z + rv.z; vz = vz > 0.f ? vz : NEG_SLOPE * vz; acc += vz * av.z;
    float vw = lv.w + rv.w; vw = vw > 0.f ? vw : NEG_SLOPE * vw; acc += vw * av.w;
  }

  __shared__ float red[256];
  red[t] = acc;
  __syncthreads();
  if (t < HEADS) {
    float s = 0.f;
#pragma unroll
    for (int k = 0; k < 16; ++k) s += red[t * 16 + k];
    logits[(size_t)e * HEADS + t] = s;
  }
}

// ---------------------------------------------------------------------------
// Kernel 3 (fused): per destination node -- softmax over its edge list,
// alpha-weighted accumulation of xl[src] rows in registers (16 float4 per
// thread spanning all 16 heads), then relu(+bias) @ W_out epilogue.
// No global FP atomics; edge-list order fixed -> deterministic.
// ---------------------------------------------------------------------------
__global__ __launch_bounds__(256)
void node_fused(const float* __restrict__ xl, const long long* __restrict__ ei,
                const int* __restrict__ rowptr, const int* __restrict__ deg,
                const int* __restrict__ ebuf, const float* __restrict__ logits,
                const float* __restrict__ bias, const float* __restrict__ Wo,
                const float* __restrict__ bo, float* __restrict__ z, int E) {
  const int n = blockIdx.x;
  const int t = threadIdx.x;
  const int start = rowptr[n];
  const int d = deg[n];                  // >= 1 (self-loop)

  __shared__ float mh[HEADS], invs[HEADS];
  __shared__ float alphaS[CHUNK][HEADS];
  __shared__ int   srcS[CHUNK];

  // ---- softmax stats per head (d is small: avg ~6) ----
  if (t < HEADS) {
    float m = -3.4e38f;
    for (int i = 0; i < d; ++i)
      m = fmaxf(m, logits[(size_t)ebuf[start + i] * HEADS + t]);
    float s = 0.f;
    for (int i = 0; i < d; ++i)
      s += expf(logits[(size_t)ebuf[start + i] * HEADS + t] - m);
    mh[t] = m;
    invs[t] = 1.f / (s + 1e-16f);
  }

  // ---- aggregation: acc[k] covers channels c = k*1024 + t*4 (head = k) ----
  float4 acc[16];
#pragma unroll
  for (int k = 0; k < 16; ++k) acc[k] = make_float4(0.f, 0.f, 0.f, 0.f);

  for (int c0 = 0; c0 < d; c0 += CHUNK) {
    const int cn = min(CHUNK, d - c0);
    __syncthreads();                     // previous chunk consumed / stats ready
    for (int q = t; q < cn * HEADS; q += 256) {
      const int i = q >> 4, h = q & 15;
      const int e = ebuf[start + c0 + i];
      alphaS[i][h] = expf(logits[(size_t)e * HEADS + h] - mh[h]) * invs[h];
    }
    if (t < cn) srcS[t] = edge_src(ei, ebuf[start + c0 + t], E);
    __syncthreads();

    for (int i = 0; i < cn; ++i) {
      const float* lrow = xl + (size_t)srcS[i] * HC;
      if (i + 1 < cn) {                  // cover next pointer-chased row
        const float* nxt = xl + (size_t)srcS[i + 1] * HC;
        __builtin_prefetch(nxt + (t << 6), 0, 0);
        __builtin_prefetch(nxt + (t << 6) + 32, 0, 0);
      }
#pragma unroll
      for (int k = 0; k < 16; ++k) {
        const float a = alphaS[i][k];
        float4 v = *(const float4*)(lrow + k * 1024 + t * 4);
        acc[k].x += a * v.x; acc[k].y += a * v.y;
        acc[k].z += a * v.z; acc[k].w += a * v.w;
      }
    }
  }

  // ---- epilogue: z[n] = relu(acc + bias) @ W_out + b_out ----
  float out[NCLS];
#pragma unroll
  for (int o = 0; o < NCLS; ++o) out[o] = 0.f;
#pragma unroll 4
  for (int k = 0; k < 16; ++k) {
    const int c = k * 1024 + t * 4;
    float4 b = *(const float4*)(bias + c);
    const float g0 = fmaxf(acc[k].x + b.x, 0.f);
    const float g1 = fmaxf(acc[k].y + b.y, 0.f);
    const float g2 = fmaxf(acc[k].z + b.z, 0.f);
    const float g3 = fmaxf(acc[k].w + b.w, 0.f);
    const float* w = Wo + (size_t)c * NCLS;
#pragma unroll
    for (int o = 0; o < NCLS; ++o)
      out[o] += g0 * w[o] + g1 * w[NCLS + o] + g2 * w[2 * NCLS + o] +
                g3 * w[3 * NCLS + o];
  }

  __shared__ float red[256 * NCLS];      // 20 KB
  __syncthreads();
#pragma unroll
  for (int o = 0; o < NCLS; ++o) red[o * 256 + t] = out[o];
  __syncthreads();
  for (int s = 128; s > 0; s >>= 1) {
    if (t < s)
#pragma unroll
      for (int o = 0; o < NCLS; ++o) red[o * 256 + t] += red[o * 256 + t + s];
    __syncthreads();
  }
  if (t < NCLS) z[(size_t)n * NCLS + t] = red[t * 256] + bo[t];
}

// ---------------------------------------------------------------------------
extern "C" void kernel_launch(void* const* d_in, const int* in_sizes, int n_in,
                              void* d_out, int out_size, void* d_ws, size_t ws_size,
                              hipStream_t stream) {
  (void)n_in; (void)out_size; (void)ws_size;
  const float*     x    = (const float*)d_in[0];
  const long long* ei   = (const long long*)d_in[1];   // int64 edge_index [2,E]
  const float*     Wl   = (const float*)d_in[2];
  const float*     Wr   = (const float*)d_in[3];
  const float*     att  = (const float*)d_in[4];
  const float*     bias = (const float*)d_in[5];
  const float*     Wo   = (const float*)d_in[6];
  const float*     bo   = (const float*)d_in[7];
  float* z = (float*)d_out;

  const int N  = in_sizes[0] / FIN;   // 4096
  const int E  = in_sizes[1] / 2;     // 16384
  const int EA = E + N;               // edges + self-loops

  // workspace: [xl][xr][logits][deg][rowptr][cursor][ebuf]
  float* xl     = (float*)d_ws;
  float* xr     = xl + (size_t)N * HC;
  float* logits = xr + (size_t)N * HC;
  int*   deg    = (int*)(logits + (size_t)EA * HEADS);
  int*   rowptr = deg + N;
  int*   cursor = rowptr + N;
  int*   ebuf   = cursor + N;

  hipMemsetAsync(deg, 0, (size_t)N * sizeof(int), stream);

  dim3 g1(2 * HC / 256, N / 16);
  gemm_xw<<<g1, 128, 0, stream>>>(x, Wl, Wr, xl, xr);

  build_deg<<<(EA + 255) / 256, 256, 0, stream>>>(ei, deg, E, EA);
  scan_deg<<<1, 256, 0, stream>>>(deg, rowptr, cursor, N);
  scatter_edges<<<(EA + 255) / 256, 256, 0, stream>>>(ei, cursor, ebuf, E, EA);
  sort_lists<<<(N + 255) / 256, 256, 0, stream>>>(rowptr, deg, ebuf, N);

  edge_logits<<<EA, 256, 0, stream>>>(xl, xr, ei, att, logits, E);

  node_fused<<<N, 256, 0, stream>>>(xl, ei, rowptr, deg, ebuf, logits,
                                    bias, Wo, bo, z, E);
}